// Loss_function_42099269435938
// MI455X (gfx1250) — compile-verified
//
#include <hip/hip_runtime.h>
#include <stdint.h>

// ---------------------------------------------------------------------------
// YOLO-style loss, shape (16384,7,7,30) fp32 -> 5 scalars.
// Pure streaming reduction: ~193 MB in, 20 B out => HBM-bound (~8.3 us floor
// at 23.3 TB/s). No GEMM structure -> WMMA not applicable; the CDNA5 win is
// the async global->LDS copy path (ASYNCcnt) + conflict-free LDS cell access.
// ---------------------------------------------------------------------------

#define TPB   128          // threads per block (4 wave32)
#define CELLS 256          // grid cells staged per block
#define NCH   30           // channels per cell
#define NB128 15           // (CELLS*NCH*4 bytes)/16 / TPB = 128-bit copies/thread/tensor

#if __has_builtin(__builtin_amdgcn_global_load_async_to_lds_b128) && \
    __has_builtin(__builtin_amdgcn_s_wait_asynccnt)
#define HAVE_ASYNC_LDS 1
typedef int v4i_t __attribute__((vector_size(16)));            // int4 (128-bit)
typedef __attribute__((address_space(1))) v4i_t gv4i_t;        // global int4
typedef __attribute__((address_space(3))) v4i_t lv4i_t;        // LDS int4
#else
#define HAVE_ASYNC_LDS 0
#endif

__device__ __forceinline__ float iou_f(float x1, float y1, float w1, float h1,
                                       float x2, float y2, float w2, float h2) {
    float l1 = x1 - 0.5f * w1, r1 = x1 + 0.5f * w1;
    float t1 = y1 - 0.5f * h1, b1 = y1 + 0.5f * h1;
    float l2 = x2 - 0.5f * w2, r2 = x2 + 0.5f * w2;
    float t2 = y2 - 0.5f * h2, b2 = y2 + 0.5f * h2;
    float in_h  = fminf(b1, b2) - fmaxf(t1, t2);
    float in_w  = fminf(r1, r2) - fmaxf(l1, l2);
    float inter = (in_h < 0.0f || in_w < 0.0f) ? 0.0f : in_h * in_w;
    float area1 = (b1 - t1) * (r1 - l1);
    float area2 = (b2 - t2) * (r2 - l2);
    return inter / (area1 + area2 - inter);
}

__global__ __launch_bounds__(TPB) void yolo_loss_main(
    const float* __restrict__ pred, const float* __restrict__ labels,
    float* __restrict__ partials, float* __restrict__ out, int use_atomic) {
    __shared__ float sp[CELLS * NCH];
    __shared__ float sl[CELLS * NCH];
    __shared__ float red[(TPB / 32) * 5];

    const int    t    = threadIdx.x;
    const size_t base = (size_t)blockIdx.x * (CELLS * NCH);

    // ---- stage 2 x 30 KB into LDS, fully coalesced ----
#if HAVE_ASYNC_LDS
    {
        const char* gp  = (const char*)(pred + base);
        const char* gl  = (const char*)(labels + base);
        char*       spc = (char*)sp;
        char*       slc = (char*)sl;
#pragma unroll
        for (int i = 0; i < NB128; ++i) {
            const int off = (t + i * TPB) * 16;
            __builtin_amdgcn_global_load_async_to_lds_b128(
                (gv4i_t*)(gp + off), (lv4i_t*)(spc + off), 0, 0);
            __builtin_amdgcn_global_load_async_to_lds_b128(
                (gv4i_t*)(gl + off), (lv4i_t*)(slc + off), 0, 0);
        }
        __builtin_amdgcn_s_wait_asynccnt(0);
    }
#else
    {
        const float4* gp  = (const float4*)(pred + base);
        const float4* gl  = (const float4*)(labels + base);
        float4*       sp4 = (float4*)sp;
        float4*       sl4 = (float4*)sl;
#pragma unroll
        for (int i = 0; i < NB128; ++i) {
            sp4[t + i * TPB] = gp[t + i * TPB];
            sl4[t + i * TPB] = gl[t + i * TPB];
        }
    }
#endif
    __syncthreads();

    // ---- per-thread compute: 2 cells each ----
    float a0 = 0.f, a1 = 0.f, a2 = 0.f, a3 = 0.f, a4 = 0.f;
#pragma unroll
    for (int cc = 0; cc < CELLS / TPB; ++cc) {
        const float* p = sp + (t + cc * TPB) * NCH;
        const float* l = sl + (t + cc * TPB) * NCH;

        const float mask = l[0];
        const float gx = l[1], gy = l[2], gw = l[3], gh = l[4];

        const float i1   = iou_f(p[1], p[2], p[3], p[4], gx, gy, gw, gh);
        const float i2   = iou_f(p[6], p[7], p[8], p[9], gx, gy, gw, gh);
        const bool  best = (i1 >= i2);
        const float sx = best ? p[1] : p[6];
        const float sy = best ? p[2] : p[7];
        const float sw = best ? p[3] : p[8];
        const float sh = best ? p[4] : p[9];
        const float imax = best ? i1 : i2;
        const float imin = best ? i2 : i1;

        const float dx = sx - gx, dy = sy - gy;
        const float center = 5.0f * (dx * dx + dy * dy);
        const float dw = sqrtf(sw) - sqrtf(gw);
        const float dh = sqrtf(sh) - sqrtf(gh);
        const float wh = 5.0f * (dw * dw + dh * dh);

        float cls = 0.0f;
#pragma unroll
        for (int c = 10; c < NCH; ++c) {
            const float d = p[c] - l[c];
            cls += d * d;
        }
        a0 += mask * center;
        a1 += mask * wh;
        a2 += mask * imax;
        a3 += mask * imin;
        a4 += mask * cls;
    }

    // ---- wave32 tree reduction, then cross-wave via LDS ----
    float v[5] = {a0, a1, a2, a3, a4};
#pragma unroll
    for (int k = 0; k < 5; ++k)
#pragma unroll
        for (int o = 16; o > 0; o >>= 1) v[k] += __shfl_down(v[k], o, 32);

    const int lane = t & 31, w = t >> 5;
    if (lane == 0) {
#pragma unroll
        for (int k = 0; k < 5; ++k) red[w * 5 + k] = v[k];
    }
    __syncthreads();
    if (t == 0) {
#pragma unroll
        for (int k = 0; k < 5; ++k) {
            float s = 0.f;
#pragma unroll
            for (int ww = 0; ww < TPB / 32; ++ww) s += red[ww * 5 + k];
            if (use_atomic) atomicAdd(&out[k], s);
            else            partials[(size_t)blockIdx.x * 5 + k] = s;
        }
    }
}

__global__ void yolo_loss_finalize(const float* __restrict__ partials,
                                   int nblocks, float* __restrict__ out) {
    __shared__ float red[256];
    for (int k = 0; k < 5; ++k) {
        float s = 0.f;
        for (int i = threadIdx.x; i < nblocks; i += 256)
            s += partials[(size_t)i * 5 + k];
        red[threadIdx.x] = s;
        __syncthreads();
        for (int off = 128; off > 0; off >>= 1) {
            if ((int)threadIdx.x < off) red[threadIdx.x] += red[threadIdx.x + off];
            __syncthreads();
        }
        if (threadIdx.x == 0) out[k] = red[0];
        __syncthreads();
    }
}

__global__ void yolo_loss_zero(float* out) {
    if (threadIdx.x < 5) out[threadIdx.x] = 0.0f;
}

extern "C" void kernel_launch(void* const* d_in, const int* in_sizes, int n_in,
                              void* d_out, int out_size, void* d_ws, size_t ws_size,
                              hipStream_t stream) {
    const float* pred   = (const float*)d_in[0];
    const float* labels = (const float*)d_in[1];
    float*       out    = (float*)d_out;

    const long long n     = in_sizes[0];          // 16384*7*7*30
    const long long cells = n / NCH;              // 802,816
    const int       nblk  = (int)(cells / CELLS); // 3136 (exact for this shape)

    const size_t need = (size_t)nblk * 5 * sizeof(float);
    if (ws_size >= need) {
        float* partials = (float*)d_ws;
        yolo_loss_main<<<nblk, TPB, 0, stream>>>(pred, labels, partials, nullptr, 0);
        yolo_loss_finalize<<<1, 256, 0, stream>>>(partials, nblk, out);
    } else {
        yolo_loss_zero<<<1, 32, 0, stream>>>(out);
        yolo_loss_main<<<nblk, TPB, 0, stream>>>(pred, labels, nullptr, out, 1);
    }
}